// SelfAttn_10127532884147
// MI455X (gfx1250) — compile-verified
//
#include <hip/hip_runtime.h>

typedef __attribute__((ext_vector_type(2))) float v2f;
typedef __attribute__((ext_vector_type(4))) float v4f;
typedef __attribute__((ext_vector_type(8))) float v8f;

constexpr int B = 4, S = 2048, H = 16, D = 64;
constexpr int NT = S / 16;            // 128 sixteen-wide tiles along sequence
constexpr float INV_TEMP = 0.125f;    // 1/sqrt(64)
constexpr float NEG_INF = -1e9f;

// One wave32 per (b, h, 16-query-row tile). Causal flash-style two-pass softmax,
// fp32 WMMA (16x16x4) for both QK^T and P*V, non-temporal stores for the 1 GiB
// attn stream so K/V stay resident in L2.
__global__ __launch_bounds__(32) void attn_fwd(const float* __restrict__ q,
                                               const float* __restrict__ k,
                                               const float* __restrict__ v,
                                               float* __restrict__ out,
                                               float* __restrict__ attn)
{
    __shared__ float plds[16 * 17];   // padded to avoid bank conflicts

    const int lane = threadIdx.x;     // 0..31
    const int half = lane >> 4;       // 0: lanes 0-15, 1: lanes 16-31
    const int l16  = lane & 15;

    const int ti = blockIdx.x % NT;   // query row tile
    const int bh = blockIdx.x / NT;
    const int h  = bh % H;
    const int b  = bh / H;

    const size_t rowstride = (size_t)H * D;                 // 1024 floats between seq rows
    const float* kb = k + ((size_t)b * S * H + h) * D;
    const float* vb = v + ((size_t)b * S * H + h) * D;

    // ---- Q tile (16x64) as 16 fp32 A-fragments, pre-scaled by 1/TEMPERATURE ----
    // A-frag layout (16x4 f32): lane holds row m=l16; lanes 0-15 -> K={0,1}, 16-31 -> K={2,3}
    const float* qp = q + (((size_t)b * S + (size_t)ti * 16 + l16) * H + h) * D + half * 2;
    v2f aQ[16];
#pragma unroll
    for (int c = 0; c < 16; ++c) {
        aQ[c].x = qp[c * 4 + 0] * INV_TEMP;
        aQ[c].y = qp[c * 4 + 1] * INV_TEMP;
    }

    // Running row stats. C-frag layout: lane holds rows m = r + half*8 (r=0..7), col n=l16.
    float rm[8], rl[8];
#pragma unroll
    for (int r = 0; r < 8; ++r) { rm[r] = -3.0e38f; rl[r] = 0.0f; }

    // ================= pass 1: row max & normalizer (lower triangle only) ==========
    for (int j = 0; j <= ti; ++j) {
        v8f c = {};
        const float* kp = kb + (size_t)(j * 16 + l16) * rowstride + half * 2;
#pragma unroll
        for (int cc = 0; cc < 16; ++cc) {
            v2f bf; bf.x = kp[cc * 4 + 0]; bf.y = kp[cc * 4 + 1];
            c = __builtin_amdgcn_wmma_f32_16x16x4_f32(false, aQ[cc], false, bf,
                                                      (short)0, c, false, false);
        }
        if (j == ti) {                 // causal mask on the diagonal tile
#pragma unroll
            for (int r = 0; r < 8; ++r)
                c[r] = (l16 > r + half * 8) ? NEG_INF : c[r];
        }
#pragma unroll
        for (int r = 0; r < 8; ++r) {
            float tm = c[r];
            tm = fmaxf(tm, __shfl_xor(tm, 1));
            tm = fmaxf(tm, __shfl_xor(tm, 2));
            tm = fmaxf(tm, __shfl_xor(tm, 4));
            tm = fmaxf(tm, __shfl_xor(tm, 8));
            const float nm = fmaxf(rm[r], tm);
            float pe = __expf(c[r] - nm);
            pe += __shfl_xor(pe, 1);
            pe += __shfl_xor(pe, 2);
            pe += __shfl_xor(pe, 4);
            pe += __shfl_xor(pe, 8);
            rl[r] = rl[r] * __expf(rm[r] - nm) + pe;
            rm[r] = nm;
        }
    }

    float il[8];
#pragma unroll
    for (int r = 0; r < 8; ++r) il[r] = 1.0f / rl[r];

    // ================= pass 2: P = softmax(S), write attn, O += P*V ================
    v8f oc[4] = {};   // O tile: 16 rows x 64 cols as 4 C-fragments
    for (int j = 0; j <= ti; ++j) {
        v8f c = {};
        const float* kp = kb + (size_t)(j * 16 + l16) * rowstride + half * 2;
#pragma unroll
        for (int cc = 0; cc < 16; ++cc) {
            v2f bf; bf.x = kp[cc * 4 + 0]; bf.y = kp[cc * 4 + 1];
            c = __builtin_amdgcn_wmma_f32_16x16x4_f32(false, aQ[cc], false, bf,
                                                      (short)0, c, false, false);
        }
        if (j == ti) {
#pragma unroll
            for (int r = 0; r < 8; ++r)
                c[r] = (l16 > r + half * 8) ? NEG_INF : c[r];
        }

        float* ap = attn + (((size_t)(b * H + h) * S + (size_t)ti * 16 + half * 8) * S)
                         + (size_t)j * 16 + l16;
#pragma unroll
        for (int r = 0; r < 8; ++r) {
            const float p = __expf(c[r] - rm[r]) * il[r];
            __builtin_nontemporal_store(p, ap + (size_t)r * S);   // streamed, bypass-ish
            plds[(r + half * 8) * 17 + l16] = p;                  // stage for re-layout
        }
        __syncthreads();   // single-wave WG: barrier is a NOP, waits settle DS ordering

        // O += P(16x16) * V(16x64), contraction over the 16 keys of this tile
#pragma unroll
        for (int kc = 0; kc < 4; ++kc) {
            v2f aP;
            aP.x = plds[l16 * 17 + kc * 4 + half * 2 + 0];
            aP.y = plds[l16 * 17 + kc * 4 + half * 2 + 1];
            const float* vp = vb + (size_t)(j * 16 + kc * 4 + half * 2) * rowstride + l16;
#pragma unroll
            for (int t = 0; t < 4; ++t) {
                v2f bV;
                bV.x = vp[(size_t)t * 16];
                bV.y = vp[rowstride + (size_t)t * 16];
                oc[t] = __builtin_amdgcn_wmma_f32_16x16x4_f32(false, aP, false, bV,
                                                              (short)0, oc[t], false, false);
            }
        }
        __syncthreads();
    }

    // ---- zero-fill the masked (upper triangle) attn columns: exact 0 after softmax ----
    const int z0 = (ti + 1) * 16;
    if (z0 < S) {
        const v4f zero = {};
        const int Z = S - z0;
        for (int m = 0; m < 16; ++m) {
            float* rp = attn + (((size_t)(b * H + h) * S + (size_t)ti * 16 + m) * S) + z0;
            for (int x = lane * 4; x < Z; x += 128)
                __builtin_nontemporal_store(zero, (v4f*)(rp + x));
        }
    }

    // ---- write O (already normalized) ----
    float* op = out + (((size_t)b * S + (size_t)ti * 16 + half * 8) * H + h) * D + l16;
#pragma unroll
    for (int t = 0; t < 4; ++t)
#pragma unroll
        for (int r = 0; r < 8; ++r)
            __builtin_nontemporal_store(oc[t][r], op + (size_t)r * rowstride + t * 16);
}

extern "C" void kernel_launch(void* const* d_in, const int* in_sizes, int n_in,
                              void* d_out, int out_size, void* d_ws, size_t ws_size,
                              hipStream_t stream) {
    const float* q = (const float*)d_in[0];
    const float* k = (const float*)d_in[1];
    const float* v = (const float*)d_in[2];
    // d_in[3] is the bool causal mask; causality is recomputed analytically.
    float* out  = (float*)d_out;
    float* attn = out + (size_t)B * S * H * D;   // outputs concatenated: (out, attn)

    dim3 grid(B * H * NT);   // 8192 single-wave workgroups
    attn_fwd<<<grid, 32, 0, stream>>>(q, k, v, out, attn);
}